// Linear_26585847562988
// MI455X (gfx1250) — compile-verified
//
#include <hip/hip_runtime.h>

typedef float v2f __attribute__((ext_vector_type(2)));
typedef float v8f __attribute__((ext_vector_type(8)));

#define IN_DIM 240

// D = A(16x4 f32) * B(4x16 f32) + C(16x16 f32)
#define WMMA_F32_4(a, b, c) \
  __builtin_amdgcn_wmma_f32_16x16x4_f32(false, (a), false, (b), (short)0, (c), false, false)

// Transposed (v-major) weight tiles in LDS, padded strides for bank-conflict-free
// b64 fragment loads: lane l reads addr v*STRIDE + k, v = vt*16 + l, k even.
// STRIDE even & STRIDE mod 64 == 2*odd  ->  banks (2l + c) mod 64 all distinct.
#define W0_STRIDE 66   // 64x64 tile
#define W1_STRIDE 34   // 32x32 tile
#define W2_STRIDE 18   // 16x16 tile

__launch_bounds__(128, 2)
__global__ void eqlin_wmma_kernel(const float* __restrict__ x,
                                  const float* __restrict__ w,
                                  float* __restrict__ out) {
  __shared__ __attribute__((aligned(16))) float lw0[64 * W0_STRIDE]; // 16.9 KB
  __shared__ __attribute__((aligned(16))) float lw1[32 * W1_STRIDE]; //  4.3 KB
  __shared__ __attribute__((aligned(16))) float lw2[16 * W2_STRIDE]; //  1.2 KB
  __shared__ __attribute__((aligned(16))) float lx[4][16 * IN_DIM];  // 4 x 15 KB

  const int tid  = threadIdx.x;
  const int lane = tid & 31;
  const int wave = tid >> 5;
  const int half = (lane >> 4) & 1;   // which 16-lane half
  const int mrow = lane & 15;         // row (A) / column (B,C) within tile

  // ---- stage weights transposed into LDS, path normalization folded in ----
  const float s0 = 0.125f;                  // 1/sqrt(64)
  const float s1 = 0.17677669529663687f;    // 1/sqrt(32)
  const float s2 = 0.25f;                   // 1/sqrt(16)
  for (int idx = tid; idx < 4096; idx += 128) {        // W0[u,v] -> lw0[v,u]
    const int u = idx >> 6, v = idx & 63;
    lw0[v * W0_STRIDE + u] = w[idx] * s0;
  }
  for (int idx = tid; idx < 1024; idx += 128) {        // W1[u,v] -> lw1[v,u]
    const int u = idx >> 5, v = idx & 31;
    lw1[v * W1_STRIDE + u] = w[4096 + idx] * s1;
  }
  for (int idx = tid; idx < 256; idx += 128) {         // W2[u,v] -> lw2[v,u]
    const int u = idx >> 4, v = idx & 15;
    lw2[v * W2_STRIDE + u] = w[5120 + idx] * s2;
  }

  // ---- stage this wave's 16x240 x tile (coalesced b128 loads) ----
  const long long n0 = (long long)blockIdx.x * 64 + wave * 16;
  const float4* xg = (const float4*)(x + n0 * IN_DIM);
  float* lxw = &lx[wave][0];
  #pragma unroll 5
  for (int t = lane; t < (16 * IN_DIM) / 4; t += 32) {
    ((float4*)lxw)[t] = xg[t];
  }
  __syncthreads();

  const v8f zero = {};
  float* outg = out + n0 * IN_DIM;
  const float* arow  = lxw + mrow * IN_DIM;          // A: this lane's row
  const float* bcol0 = lw0 + mrow * W0_STRIDE;       // B: this lane's column
  const float* bcol1 = lw1 + mrow * W1_STRIDE;
  const float* bcol2 = lw2 + mrow * W2_STRIDE;

  // ================= segment 0: [16x64] @ [64x64] =================
  {
    v8f acc0 = zero, acc1 = zero, acc2 = zero, acc3 = zero;
    for (int kk = 0; kk < 16; ++kk) {
      const int k = kk * 4 + half * 2;          // absolute K for this lane (even)
      const v2f a = *(const v2f*)(arow + k);
      acc0 = WMMA_F32_4(a, *(const v2f*)(bcol0 +                  k), acc0);
      acc1 = WMMA_F32_4(a, *(const v2f*)(bcol0 + 16 * W0_STRIDE + k), acc1);
      acc2 = WMMA_F32_4(a, *(const v2f*)(bcol0 + 32 * W0_STRIDE + k), acc2);
      acc3 = WMMA_F32_4(a, *(const v2f*)(bcol0 + 48 * W0_STRIDE + k), acc3);
    }
    #pragma unroll
    for (int j = 0; j < 8; ++j) {
      const int m = j + half * 8;               // C: VGPR j -> M=j (+8 upper half)
      float* orow = outg + m * IN_DIM;
      orow[      mrow] = acc0[j];
      orow[16 + mrow] = acc1[j];
      orow[32 + mrow] = acc2[j];
      orow[48 + mrow] = acc3[j];
    }
  }

  // ============ segment 1: 3 x ([16x32] @ [32x32]), i interleave 3 ============
  for (int i = 0; i < 3; ++i) {
    v8f acc0 = zero, acc1 = zero;
    #pragma unroll
    for (int kk = 0; kk < 8; ++kk) {
      const int k = kk * 4 + half * 2;
      v2f a;
      a.x = arow[64 + k * 3 + i];
      a.y = arow[64 + (k + 1) * 3 + i];
      acc0 = WMMA_F32_4(a, *(const v2f*)(bcol1 +                  k), acc0);
      acc1 = WMMA_F32_4(a, *(const v2f*)(bcol1 + 16 * W1_STRIDE + k), acc1);
    }
    #pragma unroll
    for (int j = 0; j < 8; ++j) {
      const int m = j + half * 8;
      float* orow = outg + m * IN_DIM;
      orow[64 +        mrow * 3 + i] = acc0[j];
      orow[64 + (16 + mrow) * 3 + i] = acc1[j];
    }
  }

  // ============ segment 2: 5 x ([16x16] @ [16x16]), i interleave 5 ============
  for (int i = 0; i < 5; ++i) {
    v8f acc = zero;
    #pragma unroll
    for (int kk = 0; kk < 4; ++kk) {
      const int k = kk * 4 + half * 2;
      v2f a;
      a.x = arow[160 + k * 5 + i];
      a.y = arow[160 + (k + 1) * 5 + i];
      acc = WMMA_F32_4(a, *(const v2f*)(bcol2 + k), acc);
    }
    #pragma unroll
    for (int j = 0; j < 8; ++j) {
      const int m = j + half * 8;
      outg[m * IN_DIM + 160 + mrow * 5 + i] = acc[j];
    }
  }
}

extern "C" void kernel_launch(void* const* d_in, const int* in_sizes, int n_in,
                              void* d_out, int out_size, void* d_ws, size_t ws_size,
                              hipStream_t stream) {
  const float* x = (const float*)d_in[0];
  const float* w = (const float*)d_in[1];
  float* out = (float*)d_out;
  const int n_rows = in_sizes[0] / IN_DIM;   // 200000, divisible by 64
  const int blocks = n_rows / 64;            // 3125 blocks x 4 waves x 16 rows
  hipLaunchKernelGGL(eqlin_wmma_kernel, dim3(blocks), dim3(128), 0, stream,
                     x, w, out);
}